// AdjacencyConvSparse_84885733638626
// MI455X (gfx1250) — compile-verified
//
#include <hip/hip_runtime.h>

typedef float v2f __attribute__((ext_vector_type(2)));
typedef float v8f __attribute__((ext_vector_type(8)));

#define N_NODES 16384
#define C_CH    64

// ---------------------------------------------------------------------------
// Zero the aggregation buffer + degree buffer (must happen every call:
// harness does not re-poison between replays, and we atomically accumulate).
// ---------------------------------------------------------------------------
__global__ void zero_kernel(float* __restrict__ p, int n) {
    int i = blockIdx.x * blockDim.x + threadIdx.x;
    if (i < n) p[i] = 0.0f;
}

// ---------------------------------------------------------------------------
// Transpose seq (C=64 x N=16384, channel-major) -> seqT (N x C, node-major)
// so each node's 64-float feature vector is a contiguous 256B block.
// Standard LDS-tiled transpose, 32x32 tiles, padded to kill bank conflicts.
// ---------------------------------------------------------------------------
__global__ __launch_bounds__(256)
void transpose_kernel(const float* __restrict__ seq, float* __restrict__ seqT) {
    __shared__ float tile[32][33];
    const int n0 = blockIdx.x * 32;           // along N (512 blocks)
    const int c0 = blockIdx.y * 32;           // along C (2 blocks)
    const int tx = threadIdx.x;               // 0..31
    const int ty = threadIdx.y;               // 0..7
#pragma unroll
    for (int i = 0; i < 32; i += 8)
        tile[ty + i][tx] = seq[(size_t)(c0 + ty + i) * N_NODES + (n0 + tx)];
    __syncthreads();
#pragma unroll
    for (int i = 0; i < 32; i += 8)
        seqT[(size_t)(n0 + ty + i) * C_CH + (c0 + tx)] = tile[tx][ty + i];
}

// ---------------------------------------------------------------------------
// Edge aggregation: one wave32 per edge (grid-stride).
//   u = idx[2e], v = idx[2e+1]
//   aggT[u,:] += seqT[v,:]   (64 floats; each lane handles a float2)
//   deg[u]    += 1.0f        (lane 0)
// seqT gather per edge is one coalesced 256B float2 load; atomic targets for
// one edge fall in 4 cachelines of L2 (everything L2-resident).
// ---------------------------------------------------------------------------
__global__ __launch_bounds__(256)
void aggregate_kernel(const int* __restrict__ idx,
                      const float* __restrict__ seqT,
                      float* __restrict__ aggT,
                      float* __restrict__ deg,
                      int M) {
    const int lane   = threadIdx.x & 31;
    const int gwave  = (int)((blockIdx.x * blockDim.x + threadIdx.x) >> 5);
    const int nwaves = (int)((gridDim.x * blockDim.x) >> 5);
    for (int e = gwave; e < M; e += nwaves) {
        const int u = idx[2 * e];
        const int v = idx[2 * e + 1];
        const float2 g = ((const float2*)(seqT + (size_t)v * C_CH))[lane];
        float* dst = aggT + (size_t)u * C_CH + 2 * lane;
        unsafeAtomicAdd(dst + 0, g.x);      // global_atomic_add_f32
        unsafeAtomicAdd(dst + 1, g.y);
        if (lane == 0) unsafeAtomicAdd(deg + u, 1.0f);
    }
}

// ---------------------------------------------------------------------------
// WMMA GEMM:  out(64 x 16384) = [W0 | W1](64x128) @ [deg .* seq ; agg](128x16384)
// v_wmma_f32_16x16x4_f32, wave32. Per block (256 threads = 8 waves):
//   wave w -> m_tile = w & 3 (16 output channels), n_sub = w >> 2 (16 nodes)
//   block covers full M=64 x N=32; 512 blocks cover N=16384.
// Fragment layouts per ISA 7.12.2:
//   A 16x4 f32 : lane m = lane&15; VGPR0 holds K = k0 + (lane<16 ? 0 : 2),
//                VGPR1 holds K+1.
//   B 4x16 f32 : lane n = lane&15; same K split across lane halves.
//   C/D 16x16  : VGPR r -> M = r + (lane<16 ? 0 : 8), N = lane&15.
// ---------------------------------------------------------------------------
__global__ __launch_bounds__(256)
void gemm_kernel(const float* __restrict__ W,     // (64, 64, 2)
                 const float* __restrict__ seq,   // (64, 16384) channel-major
                 const float* __restrict__ aggT,  // (16384, 64) node-major
                 const float* __restrict__ deg,   // (16384)
                 float* __restrict__ out) {       // (64, 16384)
    const int lane   = threadIdx.x & 31;
    const int wave   = threadIdx.x >> 5;          // 0..7
    const int m_tile = wave & 3;                  // M block (16 rows each)
    const int n_sub  = wave >> 2;                 // 0..1
    const int n_base = blockIdx.x * 32 + n_sub * 16;
    const int n      = n_base + (lane & 15);
    const int khalf  = (lane & 16) ? 2 : 0;       // K offset for this half-wave
    const int m      = m_tile * 16 + (lane & 15);

    const float degv = deg[n];
    v8f acc = {};

    // ---- K = 0..63 : A = W[:,:,0], B = deg .* seq -------------------------
#pragma unroll
    for (int k0 = 0; k0 < 64; k0 += 4) {
        const int kb = k0 + khalf;
        v2f a, b;
        a.x = W[m * 128 + kb * 2 + 0];
        a.y = W[m * 128 + (kb + 1) * 2 + 0];
        b.x = seq[(size_t)kb * N_NODES + n] * degv;
        b.y = seq[(size_t)(kb + 1) * N_NODES + n] * degv;
        acc = __builtin_amdgcn_wmma_f32_16x16x4_f32(
            false, a, false, b, (short)0, acc, false, false);
    }
    // ---- K = 64..127 : A = W[:,:,1], B = agg ------------------------------
#pragma unroll
    for (int k0 = 0; k0 < 64; k0 += 4) {
        const int kb = k0 + khalf;
        v2f a, b;
        a.x = W[m * 128 + kb * 2 + 1];
        a.y = W[m * 128 + (kb + 1) * 2 + 1];
        b.x = aggT[(size_t)n * C_CH + kb];
        b.y = aggT[(size_t)n * C_CH + kb + 1];
        acc = __builtin_amdgcn_wmma_f32_16x16x4_f32(
            false, a, false, b, (short)0, acc, false, false);
    }

    // ---- Store D: fully covers d_out (no pre-zero needed) -----------------
    const int mrow = m_tile * 16 + ((lane & 16) ? 8 : 0);
    const int ncol = n_base + (lane & 15);
#pragma unroll
    for (int r = 0; r < 8; ++r)
        out[(size_t)(mrow + r) * N_NODES + ncol] = acc[r];
}

// ---------------------------------------------------------------------------
// Workspace layout (floats):
//   [0)              aggT : 64*16384      (4 MB)
//   [64*16384)       deg  : 16384         (64 KB)
//   [64*16384+16384) seqT : 16384*64      (4 MB)
// Total ~8.06 MB.
// ---------------------------------------------------------------------------
extern "C" void kernel_launch(void* const* d_in, const int* in_sizes, int n_in,
                              void* d_out, int out_size, void* d_ws, size_t ws_size,
                              hipStream_t stream) {
    const float* seq = (const float*)d_in[0];
    const int*   idx = (const int*)d_in[1];   // JAX x64-off: int32
    const float* W   = (const float*)d_in[2];
    float*       out = (float*)d_out;

    const int M = in_sizes[1] / 2;            // number of edges

    float* aggT = (float*)d_ws;
    float* deg  = aggT + (size_t)N_NODES * C_CH;
    float* seqT = deg + N_NODES;

    // 1) zero aggregation + degree buffers
    const int zn = N_NODES * C_CH + N_NODES;
    zero_kernel<<<(zn + 255) / 256, 256, 0, stream>>>(aggT, zn);

    // 2) transpose seq -> node-major seqT
    transpose_kernel<<<dim3(N_NODES / 32, C_CH / 32), dim3(32, 8), 0, stream>>>(seq, seqT);

    // 3) edge aggregation (atomic scatter into L2-resident buffers)
    aggregate_kernel<<<2048, 256, 0, stream>>>(idx, seqT, aggT, deg, M);

    // 4) WMMA GEMM -> out
    gemm_kernel<<<N_NODES / 32, 256, 0, stream>>>(W, seq, aggT, deg, out);
}